// CausalLayerNorm_33818572488849
// MI455X (gfx1250) — compile-verified
//
#include <hip/hip_runtime.h>

// Causal layernorm, x:[8,512,8192] f32.
// Pass 1: per-(row, 256-seg) partial sums (Sx, Sx2) -> d_ws.
// Pass 2: per-(16-channel group, seg) WMMA-based tile scan:
//   inclusive cumsum along time == X @ U (U = upper-triangular ones),
//   done as 4 chained V_WMMA_F32_16X16X4_F32 with carry in the C operand.

typedef float v2f __attribute__((ext_vector_type(2)));
typedef float v8f __attribute__((ext_vector_type(8)));

#define CLN_EPS 1e-5f
#define T_DIM   8192
#define C_DIM   512
#define B_DIM   8
#define NSEG    32
#define TSEG    256              // T_DIM / NSEG
#define NROWS   (B_DIM * C_DIM)  // 4096

// Shift right by one lane within each row of 16 (DPP row_shr:1).
// Lanes 0 and 16 (invalid source) take `fill` -> fuses the shuffle with the
// tile-entry-carry select, keeping the inner loop free of LDS (ds_bpermute).
__device__ __forceinline__ float dpp_shr1_fill(float src, float fill) {
    int r = __builtin_amdgcn_update_dpp(
        __float_as_int(fill), __float_as_int(src),
        0x111 /* row_shr:1 */, 0xf, 0xf, false);
    return __int_as_float(r);
}

// ---------------- Pass 1: segment partial sums ----------------
// One wave per (row, seg). Lane loads 8 contiguous floats (2x b128),
// wave-reduces sum and sum-of-squares, lane 0 stores float2 partial.
__global__ __launch_bounds__(256) void cln_partials_kernel(
    const float* __restrict__ x, float2* __restrict__ part) {
    int gtid = blockIdx.x * blockDim.x + threadIdx.x;
    int wave = gtid >> 5;
    int lane = gtid & 31;
    int row  = wave >> 5;        // / NSEG
    int seg  = wave & (NSEG - 1);

    const float* p = x + (size_t)row * T_DIM + seg * TSEG + lane * 8;
    float4 a = ((const float4*)p)[0];
    float4 b = ((const float4*)p)[1];
    float s = a.x + a.y + a.z + a.w + b.x + b.y + b.z + b.w;
    float q = fmaf(a.x, a.x, fmaf(a.y, a.y, fmaf(a.z, a.z, a.w * a.w))) +
              fmaf(b.x, b.x, fmaf(b.y, b.y, fmaf(b.z, b.z, b.w * b.w)));
#pragma unroll
    for (int off = 16; off > 0; off >>= 1) {
        s += __shfl_xor(s, off, 32);
        q += __shfl_xor(q, off, 32);
    }
    if (lane == 0) part[row * NSEG + seg] = make_float2(s, q);
}

// ---------------- Pass 2: WMMA tile scan + normalize ----------------
// One wave per (rowgroup of 16 channels, segment of 256 timesteps).
__global__ __launch_bounds__(256) void cln_scan_kernel(
    const float* __restrict__ x, const float* __restrict__ weight,
    const float* __restrict__ bias, const float2* __restrict__ part,
    float* __restrict__ out) {
    int gtid = blockIdx.x * blockDim.x + threadIdx.x;
    int wave = gtid >> 5;
    int L    = gtid & 31;
    int rg   = wave >> 5;          // rowgroup id (0..255), 16 rows each
    int seg  = wave & (NSEG - 1);
    int row0 = rg << 4;            // base row; channel base = row0 % 512
    int c0   = row0 & (C_DIM - 1);

    int n    = L & 15;             // D-layout column (time within tile)
    int hi   = (L >> 4) & 1;       // half-wave
    int koff = hi << 1;            // A/B K interleave offset per ISA layout
    int m    = n;                  // A-layout row (channel within group)
    int rowm = row0 + m;
    const float* xrow = x + (size_t)rowm * T_DIM;

    // weight/bias in D layout: vgpr r <-> channel (hi?8:0)+r
    float wd[8], bd[8];
#pragma unroll
    for (int r = 0; r < 8; ++r) {
        int ch = c0 + (hi << 3) + r;
        wd[r] = weight[ch];
        bd[r] = bias[ch];
    }

    // Segment-entry carries: lane (.. & 15) sums partials of prior segments
    // for channel m, then scatter into D-broadcast layout (one-time bpermute).
    float cs1 = 0.f, cs2 = 0.f;
    const float2* prow = part + (size_t)rowm * NSEG;
    for (int s = 0; s < seg; ++s) {
        float2 p = prow[s];
        cs1 += p.x;
        cs2 += p.y;
    }
    v8f c1d, c2d;
#pragma unroll
    for (int r = 0; r < 8; ++r) {
        int src = (hi ? 24 : 0) + r;        // channel (hi?8:0)+r lives there
        c1d[r] = __shfl(cs1, src, 32);
        c2d[r] = __shfl(cs2, src, 32);
    }

    // Constant B operands: U[s,t] = (s <= t), rows 4k..4k+3 per K-step.
    // B 4x16 f32 layout: V0 lo=K0, V1 lo=K1, V0 hi=K2, V1 hi=K3.
    v2f bm[4];
#pragma unroll
    for (int k = 0; k < 4; ++k) {
        int s0 = 4 * k + koff;
        bm[k].x = (s0     <= n) ? 1.f : 0.f;
        bm[k].y = (s0 + 1 <= n) ? 1.f : 0.f;
    }

    int tbase = seg * TSEG;
    for (int tile = 0; tile < TSEG / 16; ++tile) {
        int t0 = tbase + tile * 16;

        // A 16x4 f32 layout: V0: K={0 lo,2 hi}, V1: K={1 lo,3 hi}
        // -> one aligned float2 per K-step per lane.
        v2f a[4], aq[4];
#pragma unroll
        for (int k = 0; k < 4; ++k) {
            a[k]  = *(const v2f*)(xrow + t0 + 4 * k + koff);
            aq[k] = a[k] * a[k];
        }

        // S1 = X @ U + carry1 ; S2 = X^2 @ U + carry2  (K-chained WMMA)
        v8f s1v = c1d, s2v = c2d;
#pragma unroll
        for (int k = 0; k < 4; ++k) {
            s1v = __builtin_amdgcn_wmma_f32_16x16x4_f32(
                false, a[k],  false, bm[k], (short)0, s1v, false, false);
            s2v = __builtin_amdgcn_wmma_f32_16x16x4_f32(
                false, aq[k], false, bm[k], (short)0, s2v, false, false);
        }

        float rc = __builtin_amdgcn_rcpf((float)(t0 + n + 1));
        v8f c1old = c1d;
#pragma unroll
        for (int r = 0; r < 8; ++r) {
            float s1   = s1v[r], s2 = s2v[r];
            float mean = s1 * rc;
            float var  = fmaf(-mean, mean, s2 * rc);
            float inv  = __builtin_amdgcn_rsqf(var + CLN_EPS);
            // recover x[t] = S1[t] - S1[t-1]; lanes n==0 pull the tile-entry
            // carry via the DPP fill operand (no LDS, no cndmask).
            float prev = dpp_shr1_fill(s1, c1old[r]);
            float xv = s1 - prev;
            float o  = fmaf((xv - mean) * inv, wd[r], bd[r]);
            out[(size_t)(row0 + (hi << 3) + r) * T_DIM + t0 + n] = o;
        }

        // new carries = column 15 of S broadcast within each half-wave
        // (lowers to v_mov_b32_dpp row_share:15)
#pragma unroll
        for (int r = 0; r < 8; ++r) {
            c1d[r] = __shfl(s1v[r], 15, 16);
            c2d[r] = __shfl(s2v[r], 15, 16);
        }
    }
}

extern "C" void kernel_launch(void* const* d_in, const int* in_sizes, int n_in,
                              void* d_out, int out_size, void* d_ws, size_t ws_size,
                              hipStream_t stream) {
    const float* x = (const float*)d_in[0];
    const float* w = (const float*)d_in[1];
    const float* b = (const float*)d_in[2];
    float* out     = (float*)d_out;
    float2* part   = (float2*)d_ws;  // 4096 * 32 * 8B = 1 MiB

    // Pass 1: 4096 rows * 32 segs = 131072 waves, 8 waves/block.
    cln_partials_kernel<<<16384, 256, 0, stream>>>(x, part);
    // Pass 2: 256 rowgroups * 32 segs = 8192 waves, 8 waves/block.
    cln_scan_kernel<<<1024, 256, 0, stream>>>(x, w, b, part, out);
    (void)in_sizes; (void)n_in; (void)out_size; (void)ws_size;
}